// RecurrentGCN_29583734735171
// MI455X (gfx1250) — compile-verified
//
#include <hip/hip_runtime.h>
#include <hip/hip_bf16.h>

// Problem sizes (fixed by the reference)
#define NN    32768      // nodes
#define UU    16384      // users
#define FIN   16
#define HL    32
#define NE    1048576    // edges

typedef __attribute__((ext_vector_type(2))) float v2f;
typedef __attribute__((ext_vector_type(8))) float v8f;

__device__ __forceinline__ float sigm(float x) { return 1.0f / (1.0f + __expf(-x)); }

// ---------------------------------------------------------------------------
// 1) xe = emb_weight[x];  m = xe @ ggc_weight.T   (ggc_weight is 16x16)
// ---------------------------------------------------------------------------
__global__ void k_embed_ggc(const int* __restrict__ x,
                            const float* __restrict__ emb,
                            const float* __restrict__ W,   // (16,16) row-major
                            float* __restrict__ xe,
                            float* __restrict__ m) {
  __shared__ float sW[256];
  for (int t = threadIdx.x; t < 256; t += blockDim.x) sW[t] = W[t];
  __syncthreads();

  int i = blockIdx.x * blockDim.x + threadIdx.x;
  if (i >= NN) return;
  int idx = x[i];
  float v[FIN];
#pragma unroll
  for (int k = 0; k < FIN; ++k) v[k] = emb[(size_t)idx * FIN + k];
#pragma unroll
  for (int k = 0; k < FIN; ++k) xe[(size_t)i * FIN + k] = v[k];
#pragma unroll
  for (int f = 0; f < FIN; ++f) {
    float acc = 0.f;
#pragma unroll
    for (int k = 0; k < FIN; ++k) acc += v[k] * sW[f * FIN + k];
    m[(size_t)i * FIN + f] = acc;
  }
}

// ---------------------------------------------------------------------------
// 2) zero scratch (agg + cnt)
// ---------------------------------------------------------------------------
__global__ void k_zero(float* __restrict__ p, int n) {
  int i = blockIdx.x * blockDim.x + threadIdx.x;
  int stride = gridDim.x * blockDim.x;
  for (; i < n; i += stride) p[i] = 0.0f;
}

// ---------------------------------------------------------------------------
// 3) edge scatter: agg[dst] += m[src]*w ; cnt[dst] += 1   (native f32 atomics)
// ---------------------------------------------------------------------------
__global__ void k_scatter(const int* __restrict__ src,
                          const int* __restrict__ dst,
                          const float* __restrict__ w,
                          const float* __restrict__ m,
                          float* __restrict__ agg,
                          float* __restrict__ cnt) {
  int j = blockIdx.x * blockDim.x + threadIdx.x;
  if (j >= NE) return;
  int s = src[j], d = dst[j];
  float wt = w[j];
  const float4* mr = (const float4*)(m + (size_t)s * FIN);
  float* ar = agg + (size_t)d * FIN;
#pragma unroll
  for (int q = 0; q < 4; ++q) {
    float4 mv = mr[q];
    unsafeAtomicAdd(ar + 4 * q + 0, mv.x * wt);
    unsafeAtomicAdd(ar + 4 * q + 1, mv.y * wt);
    unsafeAtomicAdd(ar + 4 * q + 2, mv.z * wt);
    unsafeAtomicAdd(ar + 4 * q + 3, mv.w * wt);
  }
  unsafeAtomicAdd(cnt + d, 1.0f);
}

// ---------------------------------------------------------------------------
// 4) GRU cell + LSTM cell per node; emit h_new, c_new, y=relu(h_new)
//    All weights/biases staged in LDS (7904 floats = 31.6 KB).
// ---------------------------------------------------------------------------
__global__ void k_gru_lstm(const float* __restrict__ agg,
                           const float* __restrict__ cnt,
                           const float* __restrict__ xe,
                           const float* __restrict__ gwih, const float* __restrict__ gwhh,
                           const float* __restrict__ gbih, const float* __restrict__ gbhh,
                           const float* __restrict__ lwih, const float* __restrict__ lwhh,
                           const float* __restrict__ lbih, const float* __restrict__ lbhh,
                           const float* __restrict__ h0, const float* __restrict__ c0,
                           float* __restrict__ hn_out, float* __restrict__ cn_out,
                           float* __restrict__ y_out) {
  __shared__ float s_gwih[48 * 16];
  __shared__ float s_gwhh[48 * 16];
  __shared__ float s_gb[96];            // [0..47]=b_ih, [48..95]=b_hh
  __shared__ float s_lwih[128 * 16];
  __shared__ float s_lwhh[128 * 32];
  __shared__ float s_lb[128];           // b_ih + b_hh combined

  for (int t = threadIdx.x; t < 768; t += blockDim.x) { s_gwih[t] = gwih[t]; s_gwhh[t] = gwhh[t]; }
  for (int t = threadIdx.x; t < 48;  t += blockDim.x) { s_gb[t] = gbih[t]; s_gb[48 + t] = gbhh[t]; }
  for (int t = threadIdx.x; t < 2048; t += blockDim.x) s_lwih[t] = lwih[t];
  for (int t = threadIdx.x; t < 4096; t += blockDim.x) s_lwhh[t] = lwhh[t];
  for (int t = threadIdx.x; t < 128; t += blockDim.x) s_lb[t] = lbih[t] + lbhh[t];
  __syncthreads();

  int i = blockIdx.x * blockDim.x + threadIdx.x;
  if (i >= NN) return;

  float a[FIN], xv[FIN];
  float inv = 1.0f / fmaxf(cnt[i], 1.0f);
#pragma unroll
  for (int k = 0; k < FIN; ++k) {
    a[k]  = agg[(size_t)i * FIN + k] * inv;
    xv[k] = xe[(size_t)i * FIN + k];
  }

  // GRU
  float ht[FIN];
#pragma unroll
  for (int f = 0; f < FIN; ++f) {
    float gir = s_gb[f], giz = s_gb[16 + f], gin = s_gb[32 + f];
    float ghr = s_gb[48 + f], ghz = s_gb[64 + f], ghn = s_gb[80 + f];
#pragma unroll
    for (int k = 0; k < FIN; ++k) {
      gir += a[k]  * s_gwih[f * 16 + k];
      giz += a[k]  * s_gwih[(16 + f) * 16 + k];
      gin += a[k]  * s_gwih[(32 + f) * 16 + k];
      ghr += xv[k] * s_gwhh[f * 16 + k];
      ghz += xv[k] * s_gwhh[(16 + f) * 16 + k];
      ghn += xv[k] * s_gwhh[(32 + f) * 16 + k];
    }
    float r = sigm(gir + ghr);
    float z = sigm(giz + ghz);
    float n = tanhf(gin + r * ghn);
    ht[f] = (1.0f - z) * n + z * xv[f];
  }

  // LSTM
  float h0v[HL];
#pragma unroll
  for (int k = 0; k < HL; ++k) h0v[k] = h0[(size_t)i * HL + k];

#pragma unroll 4
  for (int hh = 0; hh < HL; ++hh) {
    float gi = s_lb[hh], gf = s_lb[32 + hh], gg = s_lb[64 + hh], go = s_lb[96 + hh];
#pragma unroll
    for (int k = 0; k < FIN; ++k) {
      gi += ht[k] * s_lwih[hh * 16 + k];
      gf += ht[k] * s_lwih[(32 + hh) * 16 + k];
      gg += ht[k] * s_lwih[(64 + hh) * 16 + k];
      go += ht[k] * s_lwih[(96 + hh) * 16 + k];
    }
#pragma unroll
    for (int k = 0; k < HL; ++k) {
      gi += h0v[k] * s_lwhh[hh * 32 + k];
      gf += h0v[k] * s_lwhh[(32 + hh) * 32 + k];
      gg += h0v[k] * s_lwhh[(64 + hh) * 32 + k];
      go += h0v[k] * s_lwhh[(96 + hh) * 32 + k];
    }
    float cnew = sigm(gf) * c0[(size_t)i * HL + hh] + sigm(gi) * tanhf(gg);
    float hnew = sigm(go) * tanhf(cnew);
    cn_out[(size_t)i * HL + hh] = cnew;
    hn_out[(size_t)i * HL + hh] = hnew;
    y_out[(size_t)i * HL + hh]  = fmaxf(hnew, 0.0f);
  }
}

// ---------------------------------------------------------------------------
// 5) scores = y[:U] @ y[U:].T  — 16384x16384, K=32, f32 WMMA (16x16x4).
//    One wave per 16x64 output strip: 4 tiles, 8 chained WMMAs each.
//    Store-bandwidth bound (1 GiB out); y (4 MB) sits in L2.
// ---------------------------------------------------------------------------
__global__ void k_scores(const float* __restrict__ y, float* __restrict__ scores) {
  const int lane = threadIdx.x & 31;
  const int wave = threadIdx.x >> 5;        // 8 waves / block
  const int half = lane >> 4;               // K sub-pair selector
  const int lr   = lane & 15;               // row (A) / col (B) within tile
  const int rowTile  = blockIdx.x;          // 0..1023  -> user rows
  const int colGroup = blockIdx.y * 8 + wave; // 0..255  -> 64 item cols each

  // A fragments: user row (rowTile*16 + lr), K split across lane halves.
  const float* urow = y + (size_t)(rowTile * 16 + lr) * HL;
  v2f a[8];
#pragma unroll
  for (int s = 0; s < 8; ++s)
    a[s] = *(const v2f*)(urow + 4 * s + 2 * half);

#pragma unroll
  for (int t = 0; t < 4; ++t) {
    const int col = colGroup * 64 + t * 16;
    const float* irow = y + (size_t)(UU + col + lr) * HL;
    v8f acc = {};
#pragma unroll
    for (int s = 0; s < 8; ++s) {
      v2f b = *(const v2f*)(irow + 4 * s + 2 * half);
      acc = __builtin_amdgcn_wmma_f32_16x16x4_f32(
          /*neg_a=*/false, a[s], /*neg_b=*/false, b,
          /*c_mod=*/(short)0, acc, /*reuse_a=*/false, /*reuse_b=*/false);
    }
    float* outc = scores + col + lr;
#pragma unroll
    for (int r = 0; r < 8; ++r) {
      int row = rowTile * 16 + r + half * 8;
      outc[(size_t)row * UU] = acc[r];
    }
  }
}

// ---------------------------------------------------------------------------
extern "C" void kernel_launch(void* const* d_in, const int* in_sizes, int n_in,
                              void* d_out, int out_size, void* d_ws, size_t ws_size,
                              hipStream_t stream) {
  const int*   x    = (const int*)  d_in[0];
  const int*   e    = (const int*)  d_in[1];   // (2, E)
  const float* h    = (const float*)d_in[2];
  const float* h0   = (const float*)d_in[3];
  const float* c0   = (const float*)d_in[4];
  const float* emb  = (const float*)d_in[5];
  const float* ggc  = (const float*)d_in[6];
  const float* gwih = (const float*)d_in[7];
  const float* gwhh = (const float*)d_in[8];
  const float* gbih = (const float*)d_in[9];
  const float* gbhh = (const float*)d_in[10];
  const float* lwih = (const float*)d_in[11];
  const float* lwhh = (const float*)d_in[12];
  const float* lbih = (const float*)d_in[13];
  const float* lbhh = (const float*)d_in[14];

  float* out    = (float*)d_out;
  float* scores = out;                                  // U*U
  float* hn     = out + (size_t)UU * UU;                // N*32
  float* cn     = hn + (size_t)NN * HL;                 // N*32

  // workspace layout (floats): xe | m | agg | cnt | y   (~10.6 MB total)
  float* ws  = (float*)d_ws;
  float* xe  = ws;
  float* m   = xe  + (size_t)NN * FIN;
  float* agg = m   + (size_t)NN * FIN;
  float* cnt = agg + (size_t)NN * FIN;
  float* yv  = cnt + (size_t)NN;

  const int* esrc = e;
  const int* edst = e + NE;

  k_embed_ggc<<<NN / 256, 256, 0, stream>>>(x, emb, ggc, xe, m);

  // agg and cnt are contiguous: zero both in one pass
  k_zero<<<1024, 256, 0, stream>>>(agg, NN * FIN + NN);

  k_scatter<<<NE / 256, 256, 0, stream>>>(esrc, edst, h, m, agg, cnt);

  k_gru_lstm<<<NN / 256, 256, 0, stream>>>(agg, cnt, xe,
                                           gwih, gwhh, gbih, gbhh,
                                           lwih, lwhh, lbih, lbhh,
                                           h0, c0, hn, cn, yv);

  k_scores<<<dim3(UU / 16, UU / 64 / 8), 256, 0, stream>>>(yv, scores);
}